// VarianceApapter_89197880803890
// MI455X (gfx1250) — compile-verified
//
#include <hip/hip_runtime.h>

#define NBATCH 16
#define LSEQ   1024
#define EMB    256
#define HID    256
#define WSZ    (3 * EMB * HID)      // one conv weight tensor (floats)
#define NPAIR  (3 * 64 * 16 * 32)   // repacked pair count per conv weight
#define XPAD   260                  // LDS row stride: 16B aligned, bank-conflict free

typedef __attribute__((ext_vector_type(2))) float v2f;
typedef __attribute__((ext_vector_type(4))) float v4f;
typedef __attribute__((ext_vector_type(8))) float v8f;

__device__ __forceinline__ float wave_reduce_add(float v) {
    #pragma unroll
    for (int off = 16; off > 0; off >>= 1)
        v += __shfl_xor(v, off, 32);
    return v;
}

// Repack W[k][c][j] -> Wp[((k*64+kk4)*16+jt)*32+lane][2] so a wave's B operand for one
// 16x16x4 WMMA step is a single coalesced 256B burst of global_load_b64.
__global__ void repack_w_kernel(const float* __restrict__ W, float* __restrict__ Wp) {
    int idx = blockIdx.x * 256 + threadIdx.x;   // pair index
    if (idx >= NPAIR) return;
    int lane = idx & 31;
    int jt   = (idx >> 5) & 15;
    int kk4  = (idx >> 9) & 63;
    int k    = idx >> 15;
    int chi  = (lane >> 4) * 2;                 // hi half-lanes carry K+2 / K+3
    int j    = jt * 16 + (lane & 15);
    int c    = kk4 * 4 + chi;
    Wp[(size_t)idx * 2 + 0] = W[((size_t)(k * EMB + c)) * HID + j];
    Wp[(size_t)idx * 2 + 1] = W[((size_t)(k * EMB + c + 1)) * HID + j];
}

// Y[n,t,j] = sum_{k=0..2} sum_c X[n,t+k-1,c] * W[k,c,j] + bias[j]
// grid: (Tp/32, NBATCH), block: 512 = 16 wave32.
// Wave w: t-subtile tt=w>>3 (2 per block), 32-column group jh=w&7 (two 16x16 j-tiles).
// X tile staged via async global->LDS (ASYNCcnt); inner step: 1 LDS pair (A, shared) +
// 2 coalesced b64 (B) -> 2 WMMA (v_wmma_f32_16x16x4_f32, full fp32 precision).
__global__ void conv3_gemm_wmma(const float* __restrict__ X,
                                const float* __restrict__ Wp,
                                const float* __restrict__ bias,
                                float* __restrict__ Y,
                                int Tp, int Tvalid) {
    __shared__ float Xs[34 * XPAD];  // rows t0-1 .. t0+32
    const int n   = blockIdx.y;
    const int t0  = blockIdx.x * 32;
    const int tid = threadIdx.x;

    // Stage 34 rows x 256 floats as 16B chunks: async copy for in-range rows,
    // zero-fill for conv padding / tail rows.
    for (int ci = tid; ci < 34 * 64; ci += 512) {
        int r  = ci >> 6;            // row 0..33
        int ch = ci & 63;            // 16B chunk within row
        int gr = t0 - 1 + r;
        int fi = r * XPAD + ch * 4;  // float index into Xs (16B aligned)
        if (gr >= 0 && gr < Tvalid) {
            unsigned la = (unsigned)(uintptr_t)&Xs[fi];
            const float* gp = X + ((size_t)n * Tp + gr) * EMB + ch * 4;
            asm volatile("global_load_async_to_lds_b128 %0, %1, off"
                         :: "v"(la), "v"(gp) : "memory");
        } else {
            v4f z = {0.f, 0.f, 0.f, 0.f};
            *(v4f*)&Xs[fi] = z;
        }
    }
    asm volatile("s_wait_asynccnt 0x0" ::: "memory");
    __syncthreads();

    const int lane = tid & 31;
    const int wave = tid >> 5;
    const int tt   = wave >> 3;
    const int jh   = wave & 7;
    const int mrow = lane & 15;
    const int chi  = (lane >> 4) * 2;
    const int jt0  = jh * 2, jt1 = jh * 2 + 1;
    const v2f* W2  = (const v2f*)Wp;

    v8f acc0 = {0.f, 0.f, 0.f, 0.f, 0.f, 0.f, 0.f, 0.f};
    v8f acc1 = {0.f, 0.f, 0.f, 0.f, 0.f, 0.f, 0.f, 0.f};
    for (int k = 0; k < 3; ++k) {
        const float* Ar = &Xs[(mrow + tt * 16 + k) * XPAD];
        #pragma unroll 4
        for (int kk4 = 0; kk4 < 64; ++kk4) {
            v2f a;
            a.x = Ar[kk4 * 4 + chi];
            a.y = Ar[kk4 * 4 + chi + 1];
            int wbase = (k * 64 + kk4) * 16;
            v2f b0 = W2[(size_t)(wbase + jt0) * 32 + lane];
            v2f b1 = W2[(size_t)(wbase + jt1) * 32 + lane];
            acc0 = __builtin_amdgcn_wmma_f32_16x16x4_f32(false, a, false, b0,
                                                         (short)0, acc0, false, false);
            acc1 = __builtin_amdgcn_wmma_f32_16x16x4_f32(false, a, false, b1,
                                                         (short)0, acc1, false, false);
        }
    }
    const int jA = jh * 32 + mrow;
    const int jB = jA + 16;
    const float bA = bias[jA], bB = bias[jB];
    #pragma unroll
    for (int r = 0; r < 8; ++r) {
        int row_m = (lane < 16) ? r : r + 8;   // C/D layout: vgpr r -> M=r (lo) / r+8 (hi)
        size_t ro = ((size_t)n * Tp + t0 + tt * 16 + row_m) * HID;
        Y[ro + jA] = acc0[r] + bA;
        Y[ro + jB] = acc1[r] + bB;
    }
}

// In-place LayerNorm (biased var, eps 1e-5) + ReLU. Wave per row, block = 8 rows.
__global__ void ln_relu_kernel(float* __restrict__ H, const float* __restrict__ g,
                               const float* __restrict__ be, int rows) {
    int lane = threadIdx.x & 31;
    int row  = blockIdx.x * 8 + (threadIdx.x >> 5);
    if (row >= rows) return;
    size_t base = (size_t)row * HID + lane;
    float v[8], s = 0.f, ss = 0.f;
    #pragma unroll
    for (int i = 0; i < 8; ++i) {
        v[i] = H[base + i * 32];
        s  += v[i];
        ss += v[i] * v[i];
    }
    s  = wave_reduce_add(s);
    ss = wave_reduce_add(ss);
    float mean = s * (1.0f / HID);
    float var  = ss * (1.0f / HID) - mean * mean;
    float rs   = rsqrtf(var + 1e-5f);
    #pragma unroll
    for (int i = 0; i < 8; ++i) {
        int c = lane + i * 32;
        float o = (v[i] - mean) * rs * g[c] + be[c];
        H[base + i * 32] = fmaxf(o, 0.0f);
    }
}

// pred[row] = relu(dot(H[row,:], Wl) + bl) * alpha. Wave per row.
__global__ void linear_head_kernel(const float* __restrict__ H, const float* __restrict__ Wl,
                                   const float* __restrict__ bl, float alpha,
                                   float* __restrict__ out, int rows) {
    int lane = threadIdx.x & 31;
    int row  = blockIdx.x * 8 + (threadIdx.x >> 5);
    if (row >= rows) return;
    float s = 0.f;
    #pragma unroll
    for (int i = 0; i < 8; ++i) {
        int c = lane + i * 32;
        s += H[(size_t)row * HID + c] * Wl[c];
    }
    s = wave_reduce_add(s);
    if (lane == 0)
        out[row] = fmaxf(s + bl[0], 0.0f) * alpha;
}

// dur = int(dur_pred*2 + 0.5); inclusive scan per batch (block = 1024 threads = batch).
__global__ void dur_scan_kernel(const float* __restrict__ dp, int* __restrict__ cum,
                                int* __restrict__ tot) {
    __shared__ int s[LSEQ];
    int n = blockIdx.x, t = threadIdx.x;
    s[t] = (int)(dp[n * LSEQ + t] * 2.0f + 0.5f);
    __syncthreads();
    for (int off = 1; off < LSEQ; off <<= 1) {
        int add = (t >= off) ? s[t - off] : 0;
        __syncthreads();
        s[t] += add;
        __syncthreads();
    }
    cum[n * LSEQ + t] = s[t];
    if (t == LSEQ - 1) tot[n] = s[t];
}

// mel[n,m,:] = x[n, l(m), :] where l = smallest index with cum[l] > m; zero past totals.
__global__ void expand_kernel(const float* __restrict__ x, const int* __restrict__ cum,
                              const int* __restrict__ tot, float* __restrict__ mel, int Mp) {
    __shared__ int ls;
    int n = blockIdx.y, m = blockIdx.x, j = threadIdx.x;
    if (j == 0) {
        if (m >= tot[n]) {
            ls = -1;
        } else {
            const int* c = cum + n * LSEQ;
            int lo = 0, hi = LSEQ - 1;
            while (lo < hi) { int mid = (lo + hi) >> 1; if (c[mid] > m) hi = mid; else lo = mid + 1; }
            ls = lo;
        }
    }
    __syncthreads();
    mel[((size_t)n * Mp + m) * EMB + j] =
        (ls < 0) ? 0.0f : x[((size_t)n * LSEQ + ls) * EMB + j];
}

// searchsorted(linspace bins, v, 'left') + table gather + fused add.
__global__ void fuse_out_kernel(const float* __restrict__ mel, const float* __restrict__ pp,
                                const float* __restrict__ ep, const float* __restrict__ ptab,
                                const float* __restrict__ etab, float* __restrict__ out,
                                int M, int Mp) {
    __shared__ int sidx[2];
    int n = blockIdx.y, m = blockIdx.x, j = threadIdx.x;
    if (j == 0) {
        float p = pp[(size_t)n * Mp + m];
        float e = ep[(size_t)n * Mp + m];
        const float pstep = (800.0f - 70.0f) / 254.0f;
        const float estep = 1050.0f / 254.0f;
        int pi = 0, ei = 0;
        for (int i = 0; i < 255; ++i) {
            pi += (70.0f + (float)i * pstep) < p;
            ei += ((float)i * estep) < e;
        }
        sidx[0] = pi; sidx[1] = ei;
    }
    __syncthreads();
    size_t mo = ((size_t)n * Mp + m) * EMB + j;
    out[((size_t)n * M + m) * EMB + j] =
        mel[mo] + ptab[(size_t)sidx[0] * EMB + j] + etab[(size_t)sidx[1] * EMB + j];
}

// Remaining tuple outputs: mel_pos (1..M), dur_pred [N*L], pitch_pred [N*M], energy_pred [N*M].
__global__ void tails_kernel(const float* __restrict__ dp, const float* __restrict__ pp,
                             const float* __restrict__ ep, float* __restrict__ out,
                             int M, int Mp) {
    size_t base = (size_t)NBATCH * M * EMB;
    int idx = blockIdx.x * 256 + threadIdx.x;
    int nl = NBATCH * LSEQ;
    int nm = NBATCH * M;
    if (idx < M) {
        out[base + idx] = (float)(idx + 1);
    } else if (idx < M + nl) {
        int t = idx - M;
        out[base + M + t] = dp[t];
    } else if (idx < M + nl + nm) {
        int t = idx - M - nl;
        out[base + M + nl + t] = pp[(size_t)(t / M) * Mp + (t % M)];
    } else if (idx < M + nl + 2 * nm) {
        int t = idx - M - nl - nm;
        out[base + M + nl + nm + t] = ep[(size_t)(t / M) * Mp + (t % M)];
    }
}

static void run_predictor(const float* in, int Tp, int Tvalid,
                          const float* Wp1, const float* Wp2, const float* const* P,
                          float alpha, float* pred, float* Ha, float* Hb,
                          hipStream_t stream) {
    dim3 g(Tp / 32, NBATCH);
    int rows = NBATCH * Tp;
    conv3_gemm_wmma<<<g, 512, 0, stream>>>(in, Wp1, P[1], Ha, Tp, Tvalid);
    ln_relu_kernel<<<rows / 8, 256, 0, stream>>>(Ha, P[2], P[3], rows);
    conv3_gemm_wmma<<<g, 512, 0, stream>>>(Ha, Wp2, P[5], Hb, Tp, Tvalid);
    ln_relu_kernel<<<rows / 8, 256, 0, stream>>>(Hb, P[6], P[7], rows);
    linear_head_kernel<<<rows / 8, 256, 0, stream>>>(Hb, P[8], P[9], alpha, pred, rows);
}

extern "C" void kernel_launch(void* const* d_in, const int* in_sizes, int n_in,
                              void* d_out, int out_size, void* d_ws, size_t ws_size,
                              hipStream_t stream) {
    (void)in_sizes; (void)n_in; (void)ws_size;

    const float* x = (const float*)d_in[0];
    const float* Pd[10];  // duration params
    const float* Pp[10];  // pitch params
    const float* Pe[10];  // energy params
    for (int i = 0; i < 10; ++i) {
        Pd[i] = (const float*)d_in[1 + i];
        Pp[i] = (const float*)d_in[11 + i];
        Pe[i] = (const float*)d_in[21 + i];
    }
    const float* ptab = (const float*)d_in[31];
    const float* etab = (const float*)d_in[32];
    float* out = (float*)d_out;

    // out_size = N*M*EMB + M + N*L + 2*N*M  ->  recover M
    int M = (out_size - NBATCH * LSEQ) / (NBATCH * EMB + 2 * NBATCH + 1);
    if (M < 1) M = 1;
    int Mp   = (M + 31) & ~31;
    int Tmax = (Mp > LSEQ) ? Mp : LSEQ;
    size_t S = (size_t)NBATCH * Tmax * EMB;

    float* fw  = (float*)d_ws;
    float* wp[6];                                      // repacked conv weights
    for (int i = 0; i < 6; ++i) wp[i] = fw + (size_t)i * WSZ;
    float* Ha  = fw + 6 * (size_t)WSZ;
    float* Hb  = Ha + S;
    float* mel = Hb + S;
    float* dp  = mel + S;                              // [N*L]
    float* pp  = dp + (size_t)NBATCH * LSEQ;           // [N*Mp]
    float* ep  = pp + (size_t)NBATCH * Mp;             // [N*Mp]
    int*   cum = (int*)(ep + (size_t)NBATCH * Mp);     // [N*L]
    int*   tot = cum + NBATCH * LSEQ;                  // [N]

    // 0) repack all six conv weight tensors into WMMA operand order
    {
        const float* src[6] = { Pd[0], Pd[4], Pp[0], Pp[4], Pe[0], Pe[4] };
        int blocks = (NPAIR + 255) / 256;
        for (int i = 0; i < 6; ++i)
            repack_w_kernel<<<blocks, 256, 0, stream>>>(src[i], wp[i]);
    }

    // 1) duration predictor on x (alpha applied inside = 2.0)
    run_predictor(x, LSEQ, LSEQ, wp[0], wp[1], Pd, 2.0f, dp, Ha, Hb, stream);

    // 2) durations -> cumsum per batch
    dur_scan_kernel<<<NBATCH, LSEQ, 0, stream>>>(dp, cum, tot);

    // 3) length regulation: gather-expand into padded mel [N, Mp, EMB]
    expand_kernel<<<dim3(Mp, NBATCH), 256, 0, stream>>>(x, cum, tot, mel, Mp);

    // 4) pitch / energy predictors on mel (valid length M, padded to Mp)
    run_predictor(mel, Mp, M, wp[2], wp[3], Pp, 1.0f, pp, Ha, Hb, stream);
    run_predictor(mel, Mp, M, wp[4], wp[5], Pe, 1.0f, ep, Ha, Hb, stream);

    // 5) bucketize + embedding add -> out[0 .. N*M*EMB)
    fuse_out_kernel<<<dim3(M, NBATCH), 256, 0, stream>>>(mel, pp, ep, ptab, etab, out, M, Mp);

    // 6) mel_pos, dur_pred, pitch_pred, energy_pred tails
    int tail_total = M + NBATCH * LSEQ + 2 * NBATCH * M;
    tails_kernel<<<(tail_total + 255) / 256, 256, 0, stream>>>(dp, pp, ep, out, M, Mp);
}